// LDDMMError_77395310674027
// MI455X (gfx1250) — compile-verified
//
#include <hip/hip_runtime.h>
#include <hip/hip_bf16.h>

typedef float v2f __attribute__((ext_vector_type(2)));
typedef float v8f __attribute__((ext_vector_type(8)));

// WFLW curve tables (padded segment-slot layout).
// curves (start, n): (0,33)(33,9)(42,9)(51,9)(60,8)(68,8)(76,12)(88,10)
// npred = n-1; M = 2*npred; padded P = 16*ceil(M/16): 64,16,16,16,16,16,32,32 (total 208)
__constant__ int c_base[8] = {0, 64, 80, 96, 112, 128, 144, 176}; // padded slot base
__constant__ int c_np[8]   = {32, 8, 8, 8, 7, 7, 11, 9};          // pred segs per curve
__constant__ int c_a[8]    = {0, 33, 42, 51, 60, 68, 76, 88};     // landmark start
__constant__ int c_nt[8]   = {4, 1, 1, 1, 1, 1, 2, 2};            // 16-wide tiles per dim

#define NSLOT 208
#define WAVES_PER_BLOCK 8

__global__ __launch_bounds__(256) void lddmm_main(const float* __restrict__ pred,
                                                  const float* __restrict__ gt,
                                                  float* __restrict__ ws, int B) {
    __shared__ float4 seg[WAVES_PER_BLOCK][NSLOT];
    __shared__ float blockR[WAVES_PER_BLOCK];

    const int wave = threadIdx.x >> 5;
    const int lane = threadIdx.x & 31;
    const int b0 = blockIdx.x * WAVES_PER_BLOCK + wave;
    const bool active = (b0 < B);
    const int b = active ? b0 : (B - 1);

    const float2* p2 = reinterpret_cast<const float2*>(pred + (size_t)b * 196); // 8B aligned (784B stride)
    const float2* g2 = reinterpret_cast<const float2*>(gt   + (size_t)b * 196);

    // ---- build segment table: {cx, cy, tx, ty} per slot; zero padding ----
    for (int s = lane; s < NSLOT; s += 32) {
        int k = 0;
        while (k < 7 && s >= c_base[k + 1]) ++k;
        int off = s - c_base[k];
        int np  = c_np[k];
        float4 v = make_float4(0.f, 0.f, 0.f, 0.f);
        if (off < 2 * np) {
            bool isgt = (off >= np);
            int  i    = isgt ? (off - np) : off;
            const float2* src = isgt ? g2 : p2;
            float2 q0 = src[c_a[k] + i];
            float2 q1 = src[c_a[k] + i + 1];
            v.x = 0.5f * (q0.x + q1.x);
            v.y = 0.5f * (q0.y + q1.y);
            float tx = q1.x - q0.x, ty = q1.y - q0.y;
            if (isgt) { tx = -tx; ty = -ty; } // tau_g negated in combined set
            v.z = tx; v.w = ty;
        }
        seg[wave][s] = v;
    }
    // LDS ops are in-order per wave; subsequent same-wave reads see these writes.

    // ---- pairwise kernel sum via WMMA Gram tiles + VALU exp weights ----
    const float inv_sigma = 1.0f / 15.0f;
    const bool  loHalf = (lane < 16);
    const int   lcol   = lane & 15;
    const int   rAdd   = loHalf ? 0 : 8; // D rows: M=v (lanes 0-15), M=v+8 (lanes 16-31)
    float acc = 0.0f;

    for (int k = 0; k < 8; ++k) {
        const int base = c_base[k];
        const int nt   = c_nt[k];
        for (int ti = 0; ti < nt; ++ti) {
            const int I0 = base + ti * 16;
            // A fragment (16x4 f32): lanes 0-15 hold K=0,1; lanes 16-31 hold K=2,3 (=0 pad)
            v2f aF = {0.f, 0.f};
            if (loHalf) {
                float4 sA = seg[wave][I0 + lane];
                aF.x = sA.z; aF.y = sA.w;
            }
            for (int tj = ti; tj < nt; ++tj) {
                const int J0 = base + tj * 16;
                // B fragment (4x16 f32): lanes 0-15 hold rows K=0,1 at N=lane; 16-31 pad
                v2f bF = {0.f, 0.f};
                float4 sC = seg[wave][J0 + lcol]; // column segment (center used by all lanes)
                float cjx = sC.x, cjy = sC.y;
                if (loHalf) { bF.x = sC.z; bF.y = sC.w; }

                v8f czero = {};
                v8f G = __builtin_amdgcn_wmma_f32_16x16x4_f32(
                    false, aF, false, bF, (short)0, czero, false, false);

                const float mult = (ti == tj) ? 1.0f : 2.0f; // symmetry: off-diag doubled
                const int rbase = I0 + rAdd;
                float t = 0.0f;
#pragma unroll
                for (int v = 0; v < 8; ++v) {
                    float4 sR = seg[wave][rbase + v]; // uniform per half-wave -> LDS broadcast
                    float dx = sR.x - cjx, dy = sR.y - cjy;
                    float d2 = dx * dx + dy * dy;
                    float w  = __expf(-d2 * inv_sigma);
                    t += w * G[v];
                }
                acc += mult * t;
            }
        }
    }
#pragma unroll
    for (int m = 16; m >= 1; m >>= 1) acc += __shfl_xor(acc, m, 32);

    // ---- landmark mean error + eye normalizer ----
    float lsum = 0.0f;
    for (int i = lane; i < 98; i += 32) {
        float2 pp = p2[i], gg = g2[i];
        float dx = pp.x - gg.x, dy = pp.y - gg.y;
        lsum += sqrtf(dx * dx + dy * dy);
    }
#pragma unroll
    for (int m = 16; m >= 1; m >>= 1) lsum += __shfl_xor(lsum, m, 32);

    float2 e0 = g2[60], e1 = g2[72];
    float ex = e0.x - e1.x, ey = e0.y - e1.y;
    float eye = sqrtf(ex * ex + ey * ey);

    // r = 0.8*(lsum/98)/eye + 0.2*acc/(98*eye)
    float r = (0.8f * lsum + 0.2f * acc) / (98.0f * eye);

    if (lane == 0) blockR[wave] = active ? r : 0.0f;
    __syncthreads();
    if (threadIdx.x == 0) {
        float s = 0.0f;
        for (int w = 0; w < WAVES_PER_BLOCK; ++w) s += blockR[w]; // fixed order
        ws[blockIdx.x] = s;
    }
}

__global__ __launch_bounds__(256) void lddmm_reduce(const float* __restrict__ ws,
                                                    float* __restrict__ out,
                                                    int n, float invB) {
    __shared__ float sm[256];
    float s = 0.0f;
    for (int i = threadIdx.x; i < n; i += 256) s += ws[i]; // fixed strided order
    sm[threadIdx.x] = s;
    __syncthreads();
    for (int st = 128; st >= 1; st >>= 1) {
        if (threadIdx.x < st) sm[threadIdx.x] += sm[threadIdx.x + st];
        __syncthreads();
    }
    if (threadIdx.x == 0) out[0] = sm[0] * invB;
}

extern "C" void kernel_launch(void* const* d_in, const int* in_sizes, int n_in,
                              void* d_out, int out_size, void* d_ws, size_t ws_size,
                              hipStream_t stream) {
    const float* pred = (const float*)d_in[0];
    const float* gt   = (const float*)d_in[1];
    const int total   = in_sizes[0];
    const int B       = total / 196;
    const int nblocks = (B + WAVES_PER_BLOCK - 1) / WAVES_PER_BLOCK;
    float* ws = (float*)d_ws;

    lddmm_main<<<nblocks, 256, 0, stream>>>(pred, gt, ws, B);
    lddmm_reduce<<<1, 256, 0, stream>>>(ws, (float*)d_out, nblocks, 1.0f / (float)B);
}